// Models_18245021073832
// MI455X (gfx1250) — compile-verified
//
#include <hip/hip_runtime.h>

#define EMBED    64
#define N_USERS  100000
#define M_ITEMS  50000
#define N_NODES  (N_USERS + M_ITEMS)
#define BATCH    8192
#define N_LAYERS 3

typedef float v2f __attribute__((ext_vector_type(2)));
typedef float v8f __attribute__((ext_vector_type(8)));

// ---------------------------------------------------------------------------
// emb_cur = concat(embed_user_0, embed_item_0); emb_sum = emb_cur
// ---------------------------------------------------------------------------
__global__ void init_emb(const float* __restrict__ eu, const float* __restrict__ ei,
                         float* __restrict__ cur, float* __restrict__ sum) {
    size_t i = (size_t)blockIdx.x * blockDim.x + threadIdx.x;      // float4 index
    const size_t total  = (size_t)N_NODES * (EMBED / 4);
    const size_t userF4 = (size_t)N_USERS * (EMBED / 4);
    if (i >= total) return;
    float4 v = (i < userF4) ? ((const float4*)eu)[i]
                            : ((const float4*)ei)[i - userF4];
    ((float4*)cur)[i] = v;
    ((float4*)sum)[i] = v;
}

__global__ void zero_buf(float4* __restrict__ p, size_t n4) {
    size_t i = (size_t)blockIdx.x * blockDim.x + threadIdx.x;
    if (i < n4) p[i] = make_float4(0.f, 0.f, 0.f, 0.f);
}

__global__ void zero_scalar(float* p) { *p = 0.0f; }

// ---------------------------------------------------------------------------
// out[src] += val * in[dst]  — 16 lanes per edge, float4 gather + f32 atomics.
// Whole working set fits in the 192MB L2, so atomics resolve at L2 rate.
// ---------------------------------------------------------------------------
__global__ void spmm_scatter(const int* __restrict__ src, const int* __restrict__ dst,
                             const float* __restrict__ vals,
                             const float* __restrict__ in, float* __restrict__ out,
                             int nnz) {
    int t  = blockIdx.x * blockDim.x + threadIdx.x;
    int e  = t >> 4;
    int k4 = t & 15;
    if (e >= nnz) return;

    // Prefetch a future edge's gather row to hide random-access latency.
    int ef = e + 4096;
    if (k4 == 0 && ef < nnz)
        __builtin_prefetch(in + (size_t)dst[ef] * EMBED, 0, 1);

    int   d = dst[e];
    int   s = src[e];
    float v = vals[e];
    const float4 x = *(const float4*)(in + (size_t)d * EMBED + k4 * 4);
    float* o = out + (size_t)s * EMBED + k4 * 4;
    unsafeAtomicAdd(o + 0, v * x.x);   // global_atomic_add_f32
    unsafeAtomicAdd(o + 1, v * x.y);
    unsafeAtomicAdd(o + 2, v * x.z);
    unsafeAtomicAdd(o + 3, v * x.w);
}

__global__ void accum_sum(const float4* __restrict__ layer, float4* __restrict__ sum,
                          size_t n4) {
    size_t i = (size_t)blockIdx.x * blockDim.x + threadIdx.x;
    if (i >= n4) return;
    float4 a = sum[i], b = layer[i];
    sum[i] = make_float4(a.x + b.x, a.y + b.y, a.z + b.z, a.w + b.w);
}

// ---------------------------------------------------------------------------
// Predictions via V_WMMA_F32_16X16X4_F32: one wave handles 16 batch rows.
// D = UE_tile (16xK) x IE_tile^T (Kx16); diag(D) are the 16 dot products.
// A-operand lane map (ISA 7.12.2): lane L -> row m = L&15, k = k0 + 2*(L>=16)+v
// B-operand lane map:              lane L -> col n = L&15, same k split
// => both operands are a float2 load from this lane's own gathered row.
// light_out = emb_sum/4 folds into a final *(1/16).
// ---------------------------------------------------------------------------
__global__ void predict_wmma(const int* __restrict__ user, const int* __restrict__ item_i,
                             const int* __restrict__ item_j, const float* __restrict__ S,
                             float* __restrict__ out) {
    const int lane = threadIdx.x;      // 0..31, EXEC all ones throughout
    const int r    = lane & 15;
    const int h    = lane >> 4;
    const int b0   = blockIdx.x * 16;
    const int b    = b0 + r;

    const float* urow = S + (size_t)user[b] * EMBED;
    const float* irow = S + (size_t)(N_USERS + item_i[b]) * EMBED;
    const float* jrow = S + (size_t)(N_USERS + item_j[b]) * EMBED;

    float di = 0.f, dj = 0.f;
    int   m  = -1;

#if defined(__has_builtin) && __has_builtin(__builtin_amdgcn_wmma_f32_16x16x4_f32)
    v8f ci = {}; v8f cj = {};
#pragma unroll
    for (int k0 = 0; k0 < EMBED; k0 += 4) {
        const int k = k0 + 2 * h;
        v2f a  = *(const v2f*)(urow + k);
        v2f bi = *(const v2f*)(irow + k);
        v2f bj = *(const v2f*)(jrow + k);
        ci = __builtin_amdgcn_wmma_f32_16x16x4_f32(false, a, false, bi, (short)0, ci,
                                                   false, false);
        cj = __builtin_amdgcn_wmma_f32_16x16x4_f32(false, a, false, bj, (short)0, cj,
                                                   false, false);
    }
    // Diagonal: m<8 -> lane m, VGPR m; m>=8 -> lane m+16, VGPR m-8.
    int v = 0;
    if (h == 0 && lane < 8)  { m = lane;      v = lane;      }
    if (h == 1 && lane >= 24){ m = lane - 16; v = lane - 24; }
#pragma unroll
    for (int t = 0; t < 8; ++t)
        if (t == v) { di = ci[t]; dj = cj[t]; }
#else
    // Scalar fallback (compile-safety only): lanes 0-15 each do full dots.
    if (h == 0) {
        m = r;
        for (int k = 0; k < EMBED; ++k) {
            float a = urow[k];
            di += a * irow[k];
            dj += a * jrow[k];
        }
    }
#endif
    if (m >= 0) {
        const float sc = 1.0f / 16.0f;   // (1/4)*(1/4) from light_out normalization
        out[b0 + m]         = di * sc;
        out[BATCH + b0 + m] = dj * sc;
    }
}

// ---------------------------------------------------------------------------
// reg_loss = 0.5*(sum|u0|^2 + sum|i0|^2 + sum|j0|^2)/BATCH, one wave per elem.
// ---------------------------------------------------------------------------
__global__ void reg_loss_kernel(const int* __restrict__ user, const int* __restrict__ item_i,
                                const int* __restrict__ item_j,
                                const float* __restrict__ eu, const float* __restrict__ ei,
                                float* __restrict__ out) {
    const int lane = threadIdx.x & 31;
    const int b    = blockIdx.x * (blockDim.x / 32) + (threadIdx.x >> 5);
    if (b >= BATCH) return;

    const float2 u = *(const float2*)(eu + (size_t)user[b]   * EMBED + lane * 2);
    const float2 x = *(const float2*)(ei + (size_t)item_i[b] * EMBED + lane * 2);
    const float2 y = *(const float2*)(ei + (size_t)item_j[b] * EMBED + lane * 2);
    float s = u.x * u.x + u.y * u.y + x.x * x.x + x.y * x.y + y.x * y.x + y.y * y.y;
#pragma unroll
    for (int off = 16; off >= 1; off >>= 1) s += __shfl_xor(s, off, 32);
    if (lane == 0) unsafeAtomicAdd(out, s * (0.5f / (float)BATCH));
}

// ---------------------------------------------------------------------------
extern "C" void kernel_launch(void* const* d_in, const int* in_sizes, int n_in,
                              void* d_out, int out_size, void* d_ws, size_t ws_size,
                              hipStream_t stream) {
    const int*   user      = (const int*)d_in[0];
    const int*   item_i    = (const int*)d_in[1];
    const int*   item_j    = (const int*)d_in[2];
    const int*   edge_src  = (const int*)d_in[5];
    const int*   edge_dst  = (const int*)d_in[6];
    const float* edge_vals = (const float*)d_in[7];
    const float* eu0       = (const float*)d_in[8];
    const float* ei0       = (const float*)d_in[9];
    const int    nnz       = in_sizes[5];

    float* out = (float*)d_out;

    const size_t nodeFloats = (size_t)N_NODES * EMBED;
    float* bufA = (float*)d_ws;               // layer ping
    float* bufB = bufA + nodeFloats;          // layer pong
    float* bufS = bufB + nodeFloats;          // emb_sum accumulator

    const size_t n4  = nodeFloats / 4;
    const int    tpb = 256;
    const int    gN  = (int)((n4 + tpb - 1) / tpb);

    init_emb<<<gN, tpb, 0, stream>>>(eu0, ei0, bufA, bufS);
    zero_scalar<<<1, 1, 0, stream>>>(out + 2 * BATCH);

    float* cur = bufA;
    float* nxt = bufB;
    for (int l = 0; l < N_LAYERS; ++l) {
        zero_buf<<<gN, tpb, 0, stream>>>((float4*)nxt, n4);
        const int gS = (int)(((size_t)nnz * 16 + tpb - 1) / tpb);
        spmm_scatter<<<gS, tpb, 0, stream>>>(edge_src, edge_dst, edge_vals, cur, nxt, nnz);
        accum_sum<<<gN, tpb, 0, stream>>>((const float4*)nxt, (float4*)bufS, n4);
        float* t = cur; cur = nxt; nxt = t;
    }

    predict_wmma<<<BATCH / 16, 32, 0, stream>>>(user, item_i, item_j, bufS, out);
    reg_loss_kernel<<<BATCH / 8, 256, 0, stream>>>(user, item_i, item_j, eu0, ei0,
                                                   out + 2 * BATCH);
}